// FeatureVectorInteractions_77833397338356
// MI455X (gfx1250) — compile-verified
//
#include <hip/hip_runtime.h>

typedef float v2f __attribute__((ext_vector_type(2)));
typedef float v4f __attribute__((ext_vector_type(4)));
typedef float v8f __attribute__((ext_vector_type(8)));

#define NROW 50      // N == M == 50
#define DDIM 128
#define BPG  16      // batches per block (one WMMA tile of output rows)
#define LSTR 132     // LDS row stride in floats (pad: 132 mod 64 = 4 -> conflict-free column gathers)

__device__ __forceinline__ float pick8(v8f c, int i) {
  float r = c[0];
#pragma unroll
  for (int j = 1; j < 8; ++j) r = (i == j) ? c[j] : r;
  return r;
}

// Block = 64 threads (2 waves). Wave 0: U = rowmask-sum of V0 for 16 batches.
// Wave 1: V = colmask-sum of V1. Then wave 0 computes diag(U @ V^T) with
// 32 x V_WMMA_F32_16X16X4_F32 and writes 16 outputs.
__global__ void __launch_bounds__(64)
fvi_kernel(const int* __restrict__ VI0, const int* __restrict__ VI1,
           const float* __restrict__ V0, const float* __restrict__ V1,
           float* __restrict__ out) {
  __shared__ __align__(16) float buf[2][BPG][LSTR];  // [role][batch][d], ~16.9 KB

  const int lane = threadIdx.x & 31;
  const int role = threadIdx.x >> 5;           // 0 -> V0/U, 1 -> V1/V (wave-uniform)
  const int b0   = blockIdx.x * BPG;

  const int*   VI = role ? VI1 : VI0;
  const float* V  = role ? V1  : V0;

  // ---------- Stage 1: masked row-sums, streamed from HBM ----------
  for (int bi = 0; bi < BPG; ++bi) {
    const int b = b0 + bi;

    // masks for this batch: lane holds mask[lane] and mask[32+lane]
    const int* vi = VI + (size_t)b * NROW;
    float mlo = (vi[lane] > 0) ? 1.0f : 0.0f;
    float mhi = 0.0f;
    if (lane < NROW - 32) mhi = (vi[32 + lane] > 0) ? 1.0f : 0.0f;

    const v4f* rows = (const v4f*)(V + (size_t)b * (NROW * DDIM));

    // warm the next batch's stream (emits global_prefetch_b8)
    if (bi + 1 < BPG)
      __builtin_prefetch(V + (size_t)(b + 1) * (NROW * DDIM) + lane * 16, 0, 0);

    v4f acc = {0.0f, 0.0f, 0.0f, 0.0f};
#pragma unroll
    for (int n = 0; n < NROW; ++n) {
      // n is a compile-time constant after unroll -> v_readlane broadcast
      float m = (n < 32) ? __shfl(mlo, n, 32) : __shfl(mhi, n - 32, 32);
      v4f r = rows[n * (DDIM / 4) + lane];   // global_load_b128, fully coalesced
      acc.x = __builtin_fmaf(m, r.x, acc.x);
      acc.y = __builtin_fmaf(m, r.y, acc.y);
      acc.z = __builtin_fmaf(m, r.z, acc.z);
      acc.w = __builtin_fmaf(m, r.w, acc.w);
    }
    *(v4f*)&buf[role][bi][4 * lane] = acc;   // ds_store_b128
  }

  __syncthreads();

  // ---------- Stage 2: s[b] = diag(U @ V^T) via WMMA f32 16x16x4 ----------
  if (role == 0) {                            // wave-uniform branch; EXEC all-1s inside
    const int row = lane & 15;                // output row (batch) this lane feeds
    const int kh  = (lane >> 4) * 2;          // k sub-offset per ISA A/B 32-bit layout

    v8f c = {};
#pragma unroll
    for (int k0 = 0; k0 < DDIM; k0 += 4) {
      // A (16x4 of U) and B (4x16 of V^T) use the identical lane gather:
      //   lanes 0-15: (row, k0+0..1), lanes 16-31: (row, k0+2..3)
      v2f a  = *(const v2f*)&buf[0][row][k0 + kh];
      v2f bt = *(const v2f*)&buf[1][row][k0 + kh];
      c = __builtin_amdgcn_wmma_f32_16x16x4_f32(false, a, false, bt,
                                                (short)0, c, false, false);
    }

    // Diagonal of 16x16 C: n<8 -> VGPR n @ lane n; n>=8 -> VGPR n-8 @ lane n+16
    int n = -1;
    float d = 0.0f;
    if (lane < 8)        { n = lane;      d = pick8(c, lane);      }
    else if (lane >= 24) { n = lane - 16; d = pick8(c, lane - 24); }
    if (n >= 0) out[b0 + n] = d;
  }
}

extern "C" void kernel_launch(void* const* d_in, const int* in_sizes, int n_in,
                              void* d_out, int out_size, void* d_ws, size_t ws_size,
                              hipStream_t stream) {
  const int*   VI0 = (const int*)d_in[0];
  const int*   VI1 = (const int*)d_in[1];
  const float* V0  = (const float*)d_in[2];
  const float* V1  = (const float*)d_in[3];
  float*       out = (float*)d_out;

  const int B = in_sizes[0] / NROW;          // 16384
  dim3 grid(B / BPG);                        // 1024 blocks
  dim3 block(64);                            // 2 waves
  fvi_kernel<<<grid, block, 0, stream>>>(VI0, VI1, V0, V1, out);
}